// RNN_90460601188544
// MI455X (gfx1250) — compile-verified
//
#include <hip/hip_runtime.h>
#include <math.h>

typedef __attribute__((ext_vector_type(2))) float v2f;
typedef __attribute__((ext_vector_type(4))) float v4f;
typedef __attribute__((ext_vector_type(8))) float v8f;
typedef __attribute__((ext_vector_type(4))) int   v4i;

#define HID   128
#define IN_D  256
#define T_D   512
#define B_D   256
#define O_D   1000
#define HS    132   // 128 + 4-float pad: conflict-free strided WMMA A loads
#define XS    260   // 256 + 4-float pad: same trick for the x staging tile

// ---- gfx1250 async global->LDS copy (guarded; falls back to ld+ds_store) ----
#if defined(__AMDGCN__) && __has_builtin(__builtin_amdgcn_global_load_async_to_lds_b128)
#define USE_ASYNC_LDS 1
#endif

static __device__ __forceinline__ void async_b128(const float* g, float* l) {
#if defined(USE_ASYNC_LDS)
    __builtin_amdgcn_global_load_async_to_lds_b128(
        (__attribute__((address_space(1))) v4i*)g,
        (__attribute__((address_space(3))) v4i*)l, 0, 0);
#else
    *(v4f*)l = *(const v4f*)g;
#endif
}

static __device__ __forceinline__ void wait_async_all() {
#if defined(USE_ASYNC_LDS)
#if __has_builtin(__builtin_amdgcn_s_wait_asynccnt)
    __builtin_amdgcn_s_wait_asynccnt(0);
#else
    asm volatile("s_wait_asynccnt 0x0" ::: "memory");
#endif
#endif
}

static __device__ __forceinline__ v8f wmma_f32(v2f a, v2f b, v8f c) {
    // D = A(16x4,f32) * B(4x16,f32) + C(16x16,f32)
    return __builtin_amdgcn_wmma_f32_16x16x4_f32(
        false, a, false, b, (short)0, c, false, false);
}

// ---------------------------------------------------------------------------
// Kernel 1: xp[t][b][h] = x[b][t][:] @ W_ih^T + (b_ih + b_hh)
// 512 blocks: block bx owns timestep t = bx, sweeps 16 batch tiles.
// x tile (16x256) async-staged into double-buffered padded LDS (coalesced
// 64B/thread); W_ih slice lives in 128 VGPRs per wave; A-operands from LDS.
// ---------------------------------------------------------------------------
__global__ void __launch_bounds__(256) k_input_proj(
    const float* __restrict__ x, const float* __restrict__ W_ih,
    const float* __restrict__ b_ih, const float* __restrict__ b_hh,
    float* __restrict__ xp) {
    __shared__ float xtile[2][16 * XS];
    const int tid  = threadIdx.x;
    const int lane = tid & 31;
    const int nt   = tid >> 5;                 // wave = n-tile 0..7
    const int lrow = lane & 15;
    const int koff = (lane < 16) ? 0 : 2;
    const int mofs = (lane < 16) ? 0 : 8;
    const int n    = nt * 16 + lrow;

    const int srow = tid >> 4;                 // staging row 0..15
    const int scol = (tid & 15) * 16;          // 16 floats (64B) per thread

    v2f Breg[64];                              // W_ih B-operand in VGPRs
    const float* wrow = W_ih + (size_t)n * IN_D;
#pragma unroll
    for (int kb = 0; kb < 64; ++kb)
        Breg[kb] = *(const v2f*)(wrow + kb * 4 + koff);
    const float bias = b_ih[n] + b_hh[n];

    const int t = blockIdx.x;                  // this block's timestep

    // prologue: stage batch-tile 0 into buffer 0
    {
        const float* g = x + ((size_t)srow * T_D + t) * IN_D + scol;
        float* l = &xtile[0][srow * XS + scol];
#pragma unroll
        for (int q = 0; q < 4; ++q) async_b128(g + q * 4, l + q * 4);
    }
    wait_async_all();
    __syncthreads();

    for (int j = 0; j < 16; ++j) {
        const int buf = j & 1;
        if (j + 1 < 16) {                      // overlap next tile's fetch
            const int b0n = (j + 1) * 16;
            const float* g = x + ((size_t)(b0n + srow) * T_D + t) * IN_D + scol;
            float* l = &xtile[buf ^ 1][srow * XS + scol];
#pragma unroll
            for (int q = 0; q < 4; ++q) async_b128(g + q * 4, l + q * 4);
        }

        v8f acc = {};
#pragma unroll
        for (int kb = 0; kb < 64; ++kb) {
            v2f a = *(const v2f*)(&xtile[buf][lrow * XS + kb * 4 + koff]);
            acc = wmma_f32(a, Breg[kb], acc);
        }

        const int b0 = j * 16;
        float* orow = xp + ((size_t)t * B_D + b0) * HID + n;
#pragma unroll
        for (int v = 0; v < 8; ++v)
            orow[(size_t)(v + mofs) * HID] = acc[v] + bias;

        wait_async_all();
        __syncthreads();
    }
}

// ---------------------------------------------------------------------------
// Kernel 2: serial recurrence. 16 independent WGs (one 16-row batch tile).
// W_hh in 64 VGPRs/wave; h double-buffered in padded LDS; next step's xp tile
// async-staged into LDS while the 32-WMMA chain runs; one barrier per step.
// ---------------------------------------------------------------------------
__global__ void __launch_bounds__(256) k_recurrence(
    const float* __restrict__ xp, const float* __restrict__ W_hh,
    float* __restrict__ h_out) {
    __shared__ float hbuf[2][16 * HS];
    __shared__ float xbuf[2][16 * HS];
    const int tid  = threadIdx.x;
    const int lane = tid & 31;
    const int wave = tid >> 5;                 // n-tile 0..7
    const int b0   = blockIdx.x * 16;
    const int lrow = lane & 15;
    const int koff = (lane < 16) ? 0 : 2;
    const int mofs = (lane < 16) ? 0 : 8;
    const int n    = wave * 16 + lrow;

    const int srow = tid >> 4;                 // staging row 0..15
    const int scol = (tid & 15) * 8;           // 8 floats (32B) per thread

    v2f Breg[32];                              // W_hh B-operand in VGPRs
    const float* wrow = W_hh + (size_t)n * HID;
#pragma unroll
    for (int kb = 0; kb < 32; ++kb)
        Breg[kb] = *(const v2f*)(wrow + kb * 4 + koff);

    // h0 = 0 and stage xp tile for t=0
    for (int idx = tid; idx < 16 * HS; idx += 256)
        hbuf[0][idx] = 0.0f;
    {
        const float* g = xp + ((size_t)(b0 + srow)) * HID + scol;  // t=0
        float* l = &xbuf[0][srow * HS + scol];
        async_b128(g, l);
        async_b128(g + 4, l + 4);
    }
    wait_async_all();
    __syncthreads();

    for (int t = 0; t < T_D; ++t) {
        const int cur = t & 1;
        const int nxt = cur ^ 1;

        if (t + 1 < T_D) {                     // overlap next xp tile fetch
            const float* g =
                xp + ((size_t)(t + 1) * B_D + b0 + srow) * HID + scol;
            float* l = &xbuf[nxt][srow * HS + scol];
            async_b128(g, l);
            async_b128(g + 4, l + 4);
        }

        v8f acc = {};
#pragma unroll
        for (int kb = 0; kb < 32; ++kb) {
            v2f a = *(const v2f*)(&hbuf[cur][lrow * HS + kb * 4 + koff]);
            acc = wmma_f32(a, Breg[kb], acc);
        }

#pragma unroll
        for (int v = 0; v < 8; ++v) {
            const int m = v + mofs;
            float hv = tanhf(acc[v] + xbuf[cur][m * HS + n]);
            hbuf[nxt][m * HS + n] = hv;
        }

        wait_async_all();
        __syncthreads();
    }

    // t=511 wrote buffer 0
    for (int idx = tid; idx < 16 * HID; idx += 256) {
        const int m = idx >> 7, c = idx & 127;
        h_out[(size_t)(b0 + m) * HID + c] = hbuf[0][m * HS + c];
    }
}

// ---------------------------------------------------------------------------
// Kernel 3: out = h_last @ W_fc^T + b_fc.  [256,128]x[128,1000]
// 126 blocks x 8 waves = 1008 waves = 16 M-tiles x 63 N-tiles.
// ---------------------------------------------------------------------------
__global__ void __launch_bounds__(256) k_fc(
    const float* __restrict__ h, const float* __restrict__ W_fc,
    const float* __restrict__ b_fc, float* __restrict__ out) {
    const int tid   = threadIdx.x;
    const int lane  = tid & 31;
    const int wave  = tid >> 5;
    const int gwave = blockIdx.x * 8 + wave;   // 0..1007
    const int mt    = gwave / 63;
    const int ntile = gwave % 63;
    const int lrow  = lane & 15;
    const int koff  = (lane < 16) ? 0 : 2;
    const int mofs  = (lane < 16) ? 0 : 8;
    const int n     = ntile * 16 + lrow;
    const int nc    = (n < O_D) ? n : (O_D - 1);   // clamp OOB loads

    const float* arow = h    + (size_t)(mt * 16 + lrow) * HID + koff;
    const float* brow = W_fc + (size_t)nc * HID + koff;

    v8f acc = {};
#pragma unroll
    for (int kb = 0; kb < 32; ++kb) {
        v2f a = *(const v2f*)(arow + kb * 4);
        v2f b = *(const v2f*)(brow + kb * 4);
        acc = wmma_f32(a, b, acc);
    }

    if (n < O_D) {
        const float bias = b_fc[nc];
#pragma unroll
        for (int v = 0; v < 8; ++v) {
            const int m = mt * 16 + v + mofs;
            out[(size_t)m * O_D + n] = acc[v] + bias;
        }
    }
}

// ---------------------------------------------------------------------------
extern "C" void kernel_launch(void* const* d_in, const int* in_sizes, int n_in,
                              void* d_out, int out_size, void* d_ws, size_t ws_size,
                              hipStream_t stream) {
    const float* x    = (const float*)d_in[0];  // [256,512,256]
    const float* W_ih = (const float*)d_in[1];  // [128,256]
    const float* W_hh = (const float*)d_in[2];  // [128,128]
    const float* b_ih = (const float*)d_in[3];  // [128]
    const float* b_hh = (const float*)d_in[4];  // [128]
    const float* W_fc = (const float*)d_in[5];  // [1000,128]
    const float* b_fc = (const float*)d_in[6];  // [1000]
    float* out = (float*)d_out;                 // [256,1000]

    // Workspace: xp[T][B][H] fp32 (64 MB) + h_last[B][H] (128 KB)
    float* xp   = (float*)d_ws;
    float* h_ws = (float*)((char*)d_ws + (size_t)T_D * B_D * HID * sizeof(float));

    k_input_proj<<<T_D, 256, 0, stream>>>(x, W_ih, b_ih, b_hh, xp);
    k_recurrence<<<16, 256, 0, stream>>>(xp, W_hh, h_ws);
    k_fc<<<126, 256, 0, stream>>>(h_ws, W_fc, b_fc, out);
}